// GCNEncoder2_54606214201491
// MI455X (gfx1250) — compile-verified
//
#include <hip/hip_runtime.h>
#include <hip/hip_bf16.h>

#define HID 64
#define BN_EPS 1e-5f

typedef __attribute__((ext_vector_type(2))) float v2f;
typedef __attribute__((ext_vector_type(8))) float v8f;

static __device__ __forceinline__ void atomic_add_f32(float* p, float v) {
    __hip_atomic_fetch_add(p, v, __ATOMIC_RELAXED, __HIP_MEMORY_SCOPE_AGENT);
}

// ---------------- degree / norm ----------------
__global__ void k_deg_init(float* __restrict__ deg, int n) {
    int t = blockIdx.x * blockDim.x + threadIdx.x;
    if (t < n) deg[t] = 1.0f;  // self loop
}

__global__ void k_deg_count(const int* __restrict__ dst, float* __restrict__ deg, int e) {
    int t = blockIdx.x * blockDim.x + threadIdx.x;
    if (t < e) atomic_add_f32(deg + dst[t], 1.0f);
}

__global__ void k_deg_rsqrt(float* __restrict__ deg, int n) {
    int t = blockIdx.x * blockDim.x + threadIdx.x;
    if (t < n) deg[t] = rsqrtf(deg[t]);
}

// ---------------- WMMA GEMM: H[nrows x 64] = X[nrows x K] @ W[K x 64] ----------------
// 256 threads = 8 waves; each wave computes a 16x64 output strip via
// V_WMMA_F32_16X16X4_F32. W is staged into LDS *pre-swizzled* so that each
// lane's 2-float B fragment (rows k,k+1 of W at one column) is a contiguous
// 8-byte LDS load -> single ds_load_b64 into an aligned VGPR pair.
//
// Swizzled index: for k-step group kk = k/4, half hi = (k>>1)&1, lo = k&1,
// column c:  lw[ ((kk*2 + hi)*64 + c)*2 + lo ] = W[k*64 + c]
template <int K>
__global__ __launch_bounds__(256) void k_gemm_wmma(const float* __restrict__ X,
                                                   const float* __restrict__ W,
                                                   const float* __restrict__ bias,
                                                   float* __restrict__ H, int nrows) {
    __shared__ float lw[K * HID];
    const int tid = threadIdx.x;
    for (int i = tid; i < K * HID; i += 256) {
        int k = i >> 6;
        int c = i & 63;
        int kk = k >> 2;
        int hi = (k >> 1) & 1;
        int lo = k & 1;
        lw[(((kk * 2 + hi) * HID) + c) * 2 + lo] = W[i];
    }
    __syncthreads();

    const int wave = tid >> 5;
    const int lane = tid & 31;
    const int row0 = (blockIdx.x * 8 + wave) * 16;
    if (row0 >= nrows) return;  // wave-uniform: EXEC stays all-ones for WMMA

    const int m  = lane & 15;   // A row within tile / B column within tile
    const int hi = lane >> 4;   // K-half select (0 -> K0,K1 ; 1 -> K2,K3)

    v8f c0 = {}, c1 = {}, c2 = {}, c3 = {};
    const float* __restrict__ xrow = X + (size_t)(row0 + m) * K;
    const v2f* __restrict__ lwv = (const v2f*)lw;

    for (int k4 = 0; k4 < K / 4; ++k4) {
        v2f a = *(const v2f*)(xrow + k4 * 4 + 2 * hi);
        const v2f* __restrict__ wb = lwv + (k4 * 2 + hi) * HID + m;
        c0 = __builtin_amdgcn_wmma_f32_16x16x4_f32(false, a, false, wb[0],  (short)0, c0, false, false);
        c1 = __builtin_amdgcn_wmma_f32_16x16x4_f32(false, a, false, wb[16], (short)0, c1, false, false);
        c2 = __builtin_amdgcn_wmma_f32_16x16x4_f32(false, a, false, wb[32], (short)0, c2, false, false);
        c3 = __builtin_amdgcn_wmma_f32_16x16x4_f32(false, a, false, wb[48], (short)0, c3, false, false);
    }

    // C/D layout: VGPR i -> M = i (lanes 0-15) / M = i+8 (lanes 16-31), N = lane%16
    const int srow = row0 + hi * 8;
    float* __restrict__ hp = H + (size_t)srow * HID + m;
    const float bb0 = bias[m], bb1 = bias[16 + m], bb2 = bias[32 + m], bb3 = bias[48 + m];
#pragma unroll
    for (int i = 0; i < 8; ++i) {
        float* row = hp + (size_t)i * HID;
        row[0]  = c0[i] + bb0;
        row[16] = c1[i] + bb1;
        row[32] = c2[i] + bb2;
        row[48] = c3[i] + bb3;
    }
}

// ---------------- self-loop init: agg[i,:] = h[i,:] * dinv[i]^2 ----------------
__global__ void k_self_init(const float* __restrict__ h, const float* __restrict__ dinv,
                            float* __restrict__ out, int total) {
    int t = blockIdx.x * blockDim.x + threadIdx.x;
    if (t >= total) return;
    int r = t >> 6;
    float di = dinv[r];
    out[t] = h[t] * di * di;
}

// ---------------- edge scatter: agg[dst,:] += h[src,:] * dinv[src]*dinv[dst] --------
// 16 lanes per edge, float4 gather per lane, 4 hardware f32 atomics.
__global__ void k_edge_scatter(const float* __restrict__ h, const float* __restrict__ dinv,
                               const int* __restrict__ src, const int* __restrict__ dst,
                               float* __restrict__ out, int nedges) {
    int t = blockIdx.x * blockDim.x + threadIdx.x;
    int e = t >> 4;
    if (e >= nedges) return;
    int c = (t & 15) * 4;
    int s = src[e], d = dst[e];
    float w = dinv[s] * dinv[d];
    const float4 hv = *(const float4*)(h + (size_t)s * HID + c);
    float* o = out + (size_t)d * HID + c;
    atomic_add_f32(o + 0, hv.x * w);
    atomic_add_f32(o + 1, hv.y * w);
    atomic_add_f32(o + 2, hv.z * w);
    atomic_add_f32(o + 3, hv.w * w);
}

// ---------------- batch-norm stats: stats[0:64]=sum, stats[64:128]=sumsq ----------
__global__ void k_zero_stats(float* __restrict__ stats) {
    if (threadIdx.x < 2 * HID) stats[threadIdx.x] = 0.0f;
}

__global__ __launch_bounds__(256) void k_bn_stats(const float* __restrict__ x,
                                                  float* __restrict__ stats, int nrows) {
    __shared__ float ls[256];
    __shared__ float ls2[256];
    const int tid = threadIdx.x;
    const int c = tid & 63;
    const int rg = tid >> 6;  // 0..3
    float s = 0.0f, s2 = 0.0f;
    for (int r = blockIdx.x * 4 + rg; r < nrows; r += gridDim.x * 4) {
        float v = x[(size_t)r * HID + c];
        s += v;
        s2 += v * v;
    }
    ls[tid] = s;
    ls2[tid] = s2;
    __syncthreads();
    if (tid < HID) {
        s  = ls[tid]  + ls[tid + 64]  + ls[tid + 128]  + ls[tid + 192];
        s2 = ls2[tid] + ls2[tid + 64] + ls2[tid + 128] + ls2[tid + 192];
        atomic_add_f32(stats + tid, s);
        atomic_add_f32(stats + HID + tid, s2);
    }
}

// ---------------- batch-norm apply + ReLU ----------------
__global__ void k_bn_apply(const float* __restrict__ x, const float* __restrict__ stats,
                           const float* __restrict__ gamma, const float* __restrict__ beta,
                           float* __restrict__ y, int total, float invN) {
    int t = blockIdx.x * blockDim.x + threadIdx.x;
    if (t >= total) return;
    int c = t & 63;
    float mean = stats[c] * invN;
    float var = stats[HID + c] * invN - mean * mean;
    float sc = gamma[c] * rsqrtf(var + BN_EPS);
    float sh = beta[c] - mean * sc;
    float v = x[t] * sc + sh;
    y[t] = v > 0.0f ? v : 0.0f;
}

extern "C" void kernel_launch(void* const* d_in, const int* in_sizes, int n_in,
                              void* d_out, int out_size, void* d_ws, size_t ws_size,
                              hipStream_t stream) {
    const float* x  = (const float*)d_in[0];
    const int* ei   = (const int*)d_in[1];
    const int N = in_sizes[0] / 128;   // 50000
    const int E = in_sizes[1] / 2;     // 800000
    const int* src = ei;
    const int* dst = ei + E;

    const float* W[3] = { (const float*)d_in[2], (const float*)d_in[6], (const float*)d_in[10] };
    const float* B[3] = { (const float*)d_in[3], (const float*)d_in[7], (const float*)d_in[11] };
    const float* G[3] = { (const float*)d_in[4], (const float*)d_in[8], (const float*)d_in[12] };
    const float* BE[3] = { (const float*)d_in[5], (const float*)d_in[9], (const float*)d_in[13] };

    // workspace layout (floats)
    float* ws    = (float*)d_ws;
    float* dinv  = ws;                       // N
    float* hbuf  = dinv + N;                 // N*HID
    float* agg   = hbuf + (size_t)N * HID;   // N*HID
    float* bufX  = agg + (size_t)N * HID;    // N*HID
    float* stats = bufX + (size_t)N * HID;   // 2*HID

    const int total = N * HID;
    const float invN = 1.0f / (float)N;

    // degree -> dinv
    k_deg_init<<<(N + 255) / 256, 256, 0, stream>>>(dinv, N);
    k_deg_count<<<(E + 255) / 256, 256, 0, stream>>>(dst, dinv, E);
    k_deg_rsqrt<<<(N + 255) / 256, 256, 0, stream>>>(dinv, N);

    const float* xin = x;
    for (int l = 0; l < 3; ++l) {
        // GEMM (WMMA fp32 path)
        int gblocks = (N + 127) / 128;
        if (l == 0)
            k_gemm_wmma<128><<<gblocks, 256, 0, stream>>>(xin, W[l], B[l], hbuf, N);
        else
            k_gemm_wmma<64><<<gblocks, 256, 0, stream>>>(xin, W[l], B[l], hbuf, N);

        // aggregation: self loop + edge scatter-add
        k_self_init<<<(total + 255) / 256, 256, 0, stream>>>(hbuf, dinv, agg, total);
        long long sthreads = (long long)E * 16;
        k_edge_scatter<<<(unsigned)((sthreads + 255) / 256), 256, 0, stream>>>(
            hbuf, dinv, src, dst, agg, E);

        // batch norm + relu
        k_zero_stats<<<1, 128, 0, stream>>>(stats);
        k_bn_stats<<<256, 256, 0, stream>>>(agg, stats, N);
        float* outp = (l == 2) ? (float*)d_out : bufX;
        k_bn_apply<<<(total + 255) / 256, 256, 0, stream>>>(agg, stats, G[l], BE[l], outp,
                                                            total, invN);
        xin = bufX;
    }
}